// FlowMambaCell_54820962566298
// MI455X (gfx1250) — compile-verified
//
#include <hip/hip_runtime.h>
#include <hip/hip_bf16.h>
#include <math.h>

#define BATCH  4
#define DMODEL 64
#define DSTATE 16
#define VNUM   9
#define IMGS   32
#define HW     1024
#define MCH    96     // 64 (delta) + 16 (B) + 16 (C) output channels
#define KTOT   576    // 9 taps * 64 input channels
#define EPSV   1e-5f

typedef float v2f __attribute__((ext_vector_type(2)));
typedef float v8f __attribute__((ext_vector_type(8)));

// ---------------- Kernel 1: GroupNorm (B*GROUPS blocks), b128 streaming ----------------
__global__ void gn_kernel(const float* __restrict__ u, const float* __restrict__ gamma,
                          const float* __restrict__ beta, float* __restrict__ un) {
  int b = blockIdx.x >> 2, g = blockIdx.x & 3;
  int tid = threadIdx.x;
  const int chunk4 = 16 * HW / 4;  // 4096 float4 per (b,group)
  const float4* base4 = (const float4*)(u + (b * DMODEL + g * 16) * HW);
  float s = 0.f, sq = 0.f;
  for (int i = tid; i < chunk4; i += 256) {
    float4 x = base4[i];
    s  += x.x + x.y + x.z + x.w;
    sq += x.x * x.x + x.y * x.y + x.z * x.z + x.w * x.w;
  }
  __shared__ float ls[256], lq[256];
  ls[tid] = s; lq[tid] = sq; __syncthreads();
  for (int off = 128; off > 0; off >>= 1) {
    if (tid < off) { ls[tid] += ls[tid + off]; lq[tid] += lq[tid + off]; }
    __syncthreads();
  }
  const float cnt = 16.f * HW;
  float mu   = ls[0] / cnt;
  float var  = lq[0] / cnt - mu * mu;
  float rstd = rsqrtf(var + EPSV);
  float4* ob4 = (float4*)(un + (b * DMODEL + g * 16) * HW);
  for (int i = tid; i < chunk4; i += 256) {
    int d = g * 16 + (i >> 8);           // channel of this float4
    float sc = rstd * gamma[d], sh = beta[d];
    float4 x = base4[i], o;
    o.x = (x.x - mu) * sc + sh; o.y = (x.y - mu) * sc + sh;
    o.z = (x.z - mu) * sc + sh; o.w = (x.w - mu) * sc + sh;
    ob4[i] = o;
  }
}

// ---------------- Kernel 2: pack conv weights into 96x576 A-matrix ----------------
// Wp[oc][k], k = tap*64 + ic  (tap = dy*3+dx)
__global__ void pack_kernel(const float* __restrict__ wd, const float* __restrict__ wb,
                            const float* __restrict__ wc, float* __restrict__ Wp) {
  for (int j = blockIdx.x * blockDim.x + threadIdx.x; j < MCH * KTOT;
       j += gridDim.x * blockDim.x) {
    int oc = j / KTOT, k = j % KTOT;
    int t = k >> 6, ic = k & 63;
    float w;
    if (oc < 64)      w = wd[(oc * 64 + ic) * 9 + t];
    else if (oc < 80) w = wb[((oc - 64) * 64 + ic) * 9 + t];
    else              w = wc[((oc - 80) * 64 + ic) * 9 + t];
    Wp[j] = w;
  }
}

// ---------------- Kernel 3: implicit-GEMM conv via v_wmma_f32_16x16x4_f32 ----------------
// One wave (32 threads) per 16oc x 16pixel tile. Grid = B*32rows*6mt*2nt = 1536.
__global__ void conv_wmma_kernel(const float* __restrict__ un, const float* __restrict__ Wp,
                                 const float* __restrict__ b_delta, const float* __restrict__ dtinv,
                                 float* __restrict__ delta, float* __restrict__ Bv,
                                 float* __restrict__ Cv) {
  int blk = blockIdx.x;
  int nt = blk & 1;
  int mt = (blk >> 1) % 6;
  int y  = (blk / 12) & 31;
  int b  = blk / (12 * 32);
  int lane = threadIdx.x;
  int hi = lane >> 4, lo = lane & 15;
  int x0 = nt * 16, mbase = mt * 16;

  v8f acc = {};
  const float* unb = un + b * DMODEL * HW;
  for (int t = 0; t < 9; ++t) {
    int dy = t / 3, dx = t - dy * 3;
    int ys = (y + dy + 31) & 31;            // circular pad: input row
    int xs = (x0 + lo + dx + 31) & 31;      // circular pad: input col for this lane's N
    const float* row  = unb + ys * 32;                       // + ic*HW + xs
    const float* wrow = Wp + (mbase + lo) * KTOT + (t << 6); // this lane's M row
    for (int icb = 0; icb < 16; ++icb) {
      int k0 = icb << 2;
      v2f a, bb;
      // A 16x4 f32 layout: lanes0-15 -> {K0,K1}, lanes16-31 -> {K2,K3}, M = lane&15
      a.x = wrow[k0 + 2 * hi];
      a.y = wrow[k0 + 2 * hi + 1];
      // B 4x16 f32 layout: VGPR0 lanes0-15 K0 / lanes16-31 K1; VGPR1 K2/K3; N = lane&15
      bb.x = row[(k0 + hi)     * HW + xs];
      bb.y = row[(k0 + 2 + hi) * HW + xs];
      acc = __builtin_amdgcn_wmma_f32_16x16x4_f32(false, a, false, bb,
                                                  (short)0, acc, false, false);
    }
  }
  // C/D 16x16 f32 layout: VGPR v -> M = v + 8*(lane>=16), N = lane&15
  float dtv = dtinv[0];
  int px = x0 + lo;
  for (int v = 0; v < 8; ++v) {
    int oc = mbase + v + 8 * hi;
    float val = acc[v];
    if (oc < 64) {
      float z  = val + b_delta[oc] + dtv;
      float sp = (z > 20.f) ? z : log1pf(__expf(z));      // softplus
      sp = fminf(fmaxf(sp, 1e-4f), 5.0f);
      delta[((b * DMODEL + oc) * IMGS + y) * IMGS + px] = sp;
    } else if (oc < 80) {
      Bv[((b * DSTATE + (oc - 64)) * IMGS + y) * IMGS + px] = val;
    } else {
      Cv[((b * DSTATE + (oc - 80)) * IMGS + y) * IMGS + px] = val;
    }
  }
}

// ---------------- Kernel 4: precompute A_bar / G = delta*B_val*u_t (b128) ----------------
__global__ void ab_kernel(const float* __restrict__ delta, const float* __restrict__ Bv,
                          const float* __restrict__ u, const float* __restrict__ logA,
                          float* __restrict__ Abar, float* __restrict__ G) {
  int idx = blockIdx.x * 256 + threadIdx.x;          // float4 index
  if (idx >= BATCH * DMODEL * DSTATE * HW / 4) return;
  int p4 = idx & 255;                                 // float4 within the 1024-pixel plane
  int n  = (idx >> 8) & 15;
  int d  = (idx >> 12) & 63;
  int b  = idx >> 18;
  float4 dl = ((const float4*)delta)[(b * DMODEL + d) * 256 + p4];
  float4 bv = ((const float4*)Bv)[(b * DSTATE + n) * 256 + p4];
  float4 uu = ((const float4*)u)[(b * DMODEL + d) * 256 + p4];
  float A = -__expf(logA[d * DSTATE + n]);
  float4 ab, gg;
  ab.x = __expf(dl.x * A); ab.y = __expf(dl.y * A);
  ab.z = __expf(dl.z * A); ab.w = __expf(dl.w * A);
  gg.x = dl.x * bv.x * uu.x; gg.y = dl.y * bv.y * uu.y;
  gg.z = dl.z * bv.z * uu.z; gg.w = dl.w * bv.w * uu.w;
  ((float4*)Abar)[idx] = ab;
  ((float4*)G)[idx]    = gg;
}

// ---------------- Kernel 5: bandwidth kernel — s_new + y fused stream (b128) ----------------
// One block per (b,v,d); thread t owns pixels 4t..4t+3; n=0..15 unrolled in registers.
__global__ void stream_kernel(const float* __restrict__ s_prev, const float* __restrict__ Abar,
                              const float* __restrict__ G, const float* __restrict__ Cv,
                              const float* __restrict__ u, const float* __restrict__ Dp,
                              const int* __restrict__ gidx, float* __restrict__ y_out,
                              float* __restrict__ s_out) {
  int blk = blockIdx.x;
  int d = blk % DMODEL;
  int v = (blk / DMODEL) % VNUM;
  int b = blk / (DMODEL * VNUM);
  int tid = threadIdx.x;
  int p4 = tid * 4;                       // first pixel of this thread

  long sbase  = (long)((b * VNUM + v) * DMODEL + d) * DSTATE * HW;
  int  agbase = ((b * DMODEL + d) * DSTATE) * HW;
  const float* cvb = Cv + b * DSTATE * HW;
  float Dd = Dp[d];

  int4 ip = *(const int4*)(gidx + v * HW + p4);   // gather indices (periodic shift)
  float4 acc = {0.f, 0.f, 0.f, 0.f};

#pragma unroll
  for (int n = 0; n < DSTATE; ++n) {
    const float* sp = s_prev + sbase + (long)n * HW;
    float4 a4 = *(const float4*)(Abar + agbase + n * HW + p4);
    float4 g4 = *(const float4*)(G    + agbase + n * HW + p4);
    float4 c4 = *(const float4*)(cvb  + n * HW + p4);
    float s0 = sp[ip.x], s1 = sp[ip.y], s2 = sp[ip.z], s3 = sp[ip.w];
    float4 sn;
    sn.x = fmaf(a4.x, s0, g4.x);
    sn.y = fmaf(a4.y, s1, g4.y);
    sn.z = fmaf(a4.z, s2, g4.z);
    sn.w = fmaf(a4.w, s3, g4.w);
    *(float4*)(s_out + sbase + (long)n * HW + p4) = sn;
    acc.x = fmaf(sn.x, c4.x, acc.x);
    acc.y = fmaf(sn.y, c4.y, acc.y);
    acc.z = fmaf(sn.z, c4.z, acc.z);
    acc.w = fmaf(sn.w, c4.w, acc.w);
  }
  float4 u4 = *(const float4*)(u + (b * DMODEL + d) * HW + p4);
  float4 yv;
  yv.x = acc.x + u4.x * Dd; yv.y = acc.y + u4.y * Dd;
  yv.z = acc.z + u4.z * Dd; yv.w = acc.w + u4.w * Dd;
  *(float4*)(y_out + ((b * VNUM + v) * DMODEL + d) * HW + p4) = yv;
}

// ---------------- Launcher ----------------
extern "C" void kernel_launch(void* const* d_in, const int* in_sizes, int n_in,
                              void* d_out, int out_size, void* d_ws, size_t ws_size,
                              hipStream_t stream) {
  (void)in_sizes; (void)n_in; (void)out_size; (void)ws_size;
  const float* u_t      = (const float*)d_in[0];
  const float* s_prev   = (const float*)d_in[1];
  const float* gamma    = (const float*)d_in[2];
  const float* beta     = (const float*)d_in[3];
  const float* w_delta  = (const float*)d_in[4];
  const float* b_delta  = (const float*)d_in[5];
  const float* w_B      = (const float*)d_in[6];
  const float* w_C      = (const float*)d_in[7];
  const float* logA     = (const float*)d_in[8];
  const float* D_param  = (const float*)d_in[9];
  const float* dtinv    = (const float*)d_in[10];
  const int*   gidx     = (const int*)d_in[11];

  float* ws = (float*)d_ws;
  float* un    = ws;                 // 262144
  float* Wp    = un    + 262144;     // 55296
  float* delta = Wp    + 55296;      // 262144
  float* Bv    = delta + 262144;     // 65536
  float* Cv    = Bv    + 65536;      // 65536
  float* Abar  = Cv    + 65536;      // 4194304
  float* G     = Abar  + 4194304;    // 4194304   -> total ~36.4 MB

  float* y_out = (float*)d_out;                       // (B,V,D,H,W) = 2359296
  float* s_out = y_out + BATCH * VNUM * DMODEL * HW;  // (B,V,D,N,H,W) = 37748736

  gn_kernel<<<BATCH * 4, 256, 0, stream>>>(u_t, gamma, beta, un);
  pack_kernel<<<(MCH * KTOT + 255) / 256, 256, 0, stream>>>(w_delta, w_B, w_C, Wp);
  conv_wmma_kernel<<<BATCH * IMGS * 12, 32, 0, stream>>>(un, Wp, b_delta, dtinv,
                                                         delta, Bv, Cv);
  ab_kernel<<<(BATCH * DMODEL * DSTATE * HW / 4 + 255) / 256, 256, 0, stream>>>(
      delta, Bv, u_t, logA, Abar, G);
  stream_kernel<<<BATCH * VNUM * DMODEL, 256, 0, stream>>>(
      s_prev, Abar, G, Cv, u_t, D_param, gidx, y_out, s_out);
}